// GIN_P1_56994216018160
// MI455X (gfx1250) — compile-verified
//
#include <hip/hip_runtime.h>
#include <hip/hip_bf16.h>

typedef __attribute__((ext_vector_type(2))) float v2f;
typedef __attribute__((ext_vector_type(8))) float v8f;

#define H     128
#define HID   256
#define MBLK  256   // 8 waves * 32 rows per wave
#define NBLK  64    // columns per block (4 WMMA tiles wide)
#define BN_EPS 1e-5f

// Native no-return global fp32 atomic add (GLOBAL_ATOMIC_ADD_F32, STOREcnt-
// tracked; s_endpgm's implicit wait-idle guarantees visibility at L2 before
// the next kernel in the stream runs).
__device__ __forceinline__ void atomic_add_gl(float* p, float v) {
  asm volatile("global_atomic_add_f32 %0, %1, off" : : "v"(p), "v"(v) : "memory");
}
// LDS fp32 atomic add (lowers natively to ds_add_f32).
__device__ __forceinline__ void atomic_add_lds(float* p, float v) {
  __hip_atomic_fetch_add(p, v, __ATOMIC_RELAXED, __HIP_MEMORY_SCOPE_WORKGROUP);
}

__device__ __forceinline__ v8f wmma_f32(v2f a, v2f b, v8f c) {
  return __builtin_amdgcn_wmma_f32_16x16x4_f32(false, a, false, b, (short)0, c,
                                               false, false);
}

// Stage W[n][k] (row-major [Nc,K]) transposed into LDS as lds[g][n][kk],
// g = k/4, kk = k%4, n in [0,NBLK). float4 global loads, float4 LDS stores.
template <int K>
__device__ __forceinline__ void stage_w(const float* __restrict__ W, int nblk0,
                                        float* lds_w, int tid) {
  constexpr int G = K / 4;
  for (int i = tid; i < NBLK * G; i += 256) {
    const int n = i / G;
    const int g = i % G;
    const float4 wv = *(const float4*)(W + (size_t)(nblk0 + n) * K + 4 * g);
    *(float4*)(lds_w + ((size_t)g * NBLK + n) * 4) = wv;
  }
}

// Per-wave 32(M) x 64(N) tile mainloop over K in steps of 4.
// Rows are clamped to M-1 (branch-free): out-of-range rows compute garbage
// that is never stored and never enters the BN statistics.
template <int K>
__device__ __forceinline__ void mainloop(const float* __restrict__ A, int M,
                                         int mbase, const float* lds_w,
                                         int lane, v8f acc[2][4]) {
  const int nloc = lane & 15;
  const int koff = (lane >> 4) * 2;  // lanes 16-31 hold k+2,k+3
  const int m0 = min(mbase + nloc, M - 1);
  const int m1 = min(mbase + 16 + nloc, M - 1);
  const float* arow0 = A + (size_t)m0 * K + koff;
  const float* arow1 = A + (size_t)m1 * K + koff;
  const float* bbase = lds_w + (size_t)nloc * 4 + koff;
#pragma unroll 4
  for (int k0 = 0; k0 < K; k0 += 4) {
    const v2f a0 = *(const v2f*)(arow0 + k0);
    const v2f a1 = *(const v2f*)(arow1 + k0);
    const float* bp = bbase + (size_t)(k0 >> 2) * (NBLK * 4);
#pragma unroll
    for (int t = 0; t < 4; ++t) {
      const v2f b = *(const v2f*)(bp + t * 64);  // t*16 columns * 4 floats
      acc[0][t] = wmma_f32(a0, b, acc[0][t]);
      acc[1][t] = wmma_f32(a1, b, acc[1][t]);
    }
  }
}

__global__ void fill_zero_kernel(float* __restrict__ p, size_t n) {
  for (size_t i = (size_t)blockIdx.x * blockDim.x + threadIdx.x; i < n;
       i += (size_t)gridDim.x * blockDim.x)
    p[i] = 0.f;
}

// One edge handled by 32 lane-slots of 4 channels: float4 gather of x[src],
// 4 native global f32 atomic adds into agg[dst] (lives in L2: 51 MB).
__global__ void scatter_kernel(const float* __restrict__ x,
                               const long long* __restrict__ ei,
                               float* __restrict__ agg, long long E) {
  const long long tid = (long long)blockIdx.x * blockDim.x + threadIdx.x;
  const long long e = tid >> 5;
  if (e >= E) return;
  const int c = ((int)tid & 31) * 4;
  const long long dst = ei[e];
  const long long src = ei[E + e];
  const float4 v = *(const float4*)(x + (size_t)src * H + c);
  float* ap = agg + (size_t)dst * H + c;
  atomic_add_gl(ap + 0, v.x);
  atomic_add_gl(ap + 1, v.y);
  atomic_add_gl(ap + 2, v.z);
  atomic_add_gl(ap + 3, v.w);
}

// h = x @ lin_w^T + lin_b + (1+eps)*[x,x] + [agg,agg]   (K=128, Nc=256)
__global__ void __launch_bounds__(256)
gemm_fuse_kernel(const float* __restrict__ x, const float* __restrict__ agg,
                 const float* __restrict__ lin_w, const float* __restrict__ lin_b,
                 const float* __restrict__ eps_p, float* __restrict__ out, int M) {
  __shared__ float lds_w[(H / 4) * NBLK * 4];  // 32 KB
  const int tid = threadIdx.x;
  const int nblk0 = blockIdx.y * NBLK;
  stage_w<H>(lin_w, nblk0, lds_w, tid);
  __syncthreads();

  const int lane  = tid & 31;
  const int mbase = blockIdx.x * MBLK + (tid >> 5) * 32;
  v8f acc[2][4] = {};
  mainloop<H>(x, M, mbase, lds_w, lane, acc);

  const float opeps = 1.0f + eps_p[0];
  const int mhalf = (lane >> 4) * 8;
  const int nloc  = lane & 15;
#pragma unroll
  for (int t = 0; t < 4; ++t) {
    const int n = nblk0 + t * 16 + nloc;
    const int c = n & (H - 1);
    const float lb = lin_b[n];
#pragma unroll
    for (int mt = 0; mt < 2; ++mt) {
#pragma unroll
      for (int r = 0; r < 8; ++r) {
        const int m = mbase + mt * 16 + mhalf + r;
        if (m < M) {
          const float xs = x[(size_t)m * H + c];
          const float ag = agg[(size_t)m * H + c];
          out[(size_t)m * HID + n] = acc[mt][t][r] + lb + opeps * xs + ag;
        }
      }
    }
  }
}

// out = A @ W^T (K=256, Nc=256) + per-column sum / sum-of-squares into stats.
__global__ void __launch_bounds__(256)
gemm_bn_kernel(const float* __restrict__ A, const float* __restrict__ W,
               float* __restrict__ out, float* __restrict__ stats, int M) {
  __shared__ float lds_w[(HID / 4) * NBLK * 4];  // 64 KB
  __shared__ float s_sum[NBLK];
  __shared__ float s_sq[NBLK];
  const int tid = threadIdx.x;
  if (tid < NBLK) { s_sum[tid] = 0.f; s_sq[tid] = 0.f; }
  const int nblk0 = blockIdx.y * NBLK;
  stage_w<HID>(W, nblk0, lds_w, tid);
  __syncthreads();

  const int lane  = tid & 31;
  const int mbase = blockIdx.x * MBLK + (tid >> 5) * 32;
  v8f acc[2][4] = {};
  mainloop<HID>(A, M, mbase, lds_w, lane, acc);

  const int mhalf = (lane >> 4) * 8;
  const int nloc  = lane & 15;
#pragma unroll
  for (int t = 0; t < 4; ++t) {
    const int n = nblk0 + t * 16 + nloc;
    float csum = 0.f, csq = 0.f;
#pragma unroll
    for (int mt = 0; mt < 2; ++mt) {
#pragma unroll
      for (int r = 0; r < 8; ++r) {
        const int m = mbase + mt * 16 + mhalf + r;
        if (m < M) {
          const float v = acc[mt][t][r];
          out[(size_t)m * HID + n] = v;
          csum += v;
          csq  += v * v;
        }
      }
    }
    // lanes L and L+16 hold the two row-halves of the SAME column n: fold them
    csum += __shfl_xor(csum, 16);
    csq  += __shfl_xor(csq, 16);
    if (lane < 16) {
      atomic_add_lds(&s_sum[t * 16 + nloc], csum);
      atomic_add_lds(&s_sq[t * 16 + nloc], csq);
    }
  }
  __syncthreads();
  if (tid < NBLK) {
    atomic_add_gl(&stats[nblk0 + tid], s_sum[tid]);
    atomic_add_gl(&stats[HID + nblk0 + tid], s_sq[tid]);
  }
}

// In-place BN (training-mode stats, biased variance) + ReLU over [M,HID].
__global__ void bn_relu_kernel(float* __restrict__ t,
                               const float* __restrict__ gamma,
                               const float* __restrict__ beta,
                               const float* __restrict__ stats, int M) {
  const float invN = 1.0f / (float)M;
  const size_t total = (size_t)M * HID;
  for (size_t i = (size_t)blockIdx.x * blockDim.x + threadIdx.x; i < total;
       i += (size_t)gridDim.x * blockDim.x) {
    const int j = (int)(i & (HID - 1));
    const float mean = stats[j] * invN;
    const float var  = stats[HID + j] * invN - mean * mean;
    const float sc   = gamma[j] * rsqrtf(var + BN_EPS);
    const float sh   = beta[j] - mean * sc;
    t[i] = fmaxf(0.f, t[i] * sc + sh);
  }
}

// out = A @ w3^T + b3   (K=256, Nc=128)
__global__ void __launch_bounds__(256)
gemm_out_kernel(const float* __restrict__ A, const float* __restrict__ W,
                const float* __restrict__ b3, float* __restrict__ out, int M) {
  __shared__ float lds_w[(HID / 4) * NBLK * 4];  // 64 KB
  const int tid = threadIdx.x;
  const int nblk0 = blockIdx.y * NBLK;
  stage_w<HID>(W, nblk0, lds_w, tid);
  __syncthreads();

  const int lane  = tid & 31;
  const int mbase = blockIdx.x * MBLK + (tid >> 5) * 32;
  v8f acc[2][4] = {};
  mainloop<HID>(A, M, mbase, lds_w, lane, acc);

  const int mhalf = (lane >> 4) * 8;
  const int nloc  = lane & 15;
#pragma unroll
  for (int t = 0; t < 4; ++t) {
    const int n = nblk0 + t * 16 + nloc;
    const float bb = b3[n];
#pragma unroll
    for (int mt = 0; mt < 2; ++mt) {
#pragma unroll
      for (int r = 0; r < 8; ++r) {
        const int m = mbase + mt * 16 + mhalf + r;
        if (m < M) out[(size_t)m * H + n] = acc[mt][t][r] + bb;
      }
    }
  }
}

extern "C" void kernel_launch(void* const* d_in, const int* in_sizes, int n_in,
                              void* d_out, int out_size, void* d_ws, size_t ws_size,
                              hipStream_t stream) {
  const float*     x     = (const float*)d_in[0];
  const long long* ei    = (const long long*)d_in[1];
  const float*     lin_w = (const float*)d_in[2];
  const float*     lin_b = (const float*)d_in[3];
  const float*     eps   = (const float*)d_in[4];
  const float*     w1    = (const float*)d_in[5];
  const float*     g1    = (const float*)d_in[6];
  const float*     b1    = (const float*)d_in[7];
  const float*     w2    = (const float*)d_in[8];
  const float*     g2    = (const float*)d_in[9];
  const float*     b2    = (const float*)d_in[10];
  const float*     w3    = (const float*)d_in[11];
  const float*     b3    = (const float*)d_in[12];
  float*           out   = (float*)d_out;

  const int       N    = in_sizes[0] / H;
  const long long E    = (long long)(in_sizes[1] / 2);
  const int       Mpad = ((N + MBLK - 1) / MBLK) * MBLK;

  float* agg   = (float*)d_ws;                       // [Mpad,128]
  float* hbuf  = agg  + (size_t)Mpad * H;            // [Mpad,256]
  float* tbuf  = hbuf + (size_t)Mpad * HID;          // [Mpad,256]
  float* stats = tbuf + (size_t)Mpad * HID;          // sum1,sq1,sum2,sq2 (4*HID)

  // Accumulators must be re-zeroed every call (graph replays).
  fill_zero_kernel<<<2048, 256, 0, stream>>>(agg, (size_t)Mpad * H);
  fill_zero_kernel<<<4, 256, 0, stream>>>(stats, 4 * HID);

  const long long sthreads = E * 32;
  scatter_kernel<<<(int)((sthreads + 255) / 256), 256, 0, stream>>>(x, ei, agg, E);

  dim3 gmm(Mpad / MBLK, HID / NBLK);
  gemm_fuse_kernel<<<gmm, 256, 0, stream>>>(x, agg, lin_w, lin_b, eps, hbuf, N);

  gemm_bn_kernel<<<gmm, 256, 0, stream>>>(hbuf, w1, tbuf, stats, N);
  bn_relu_kernel<<<2048, 256, 0, stream>>>(tbuf, g1, b1, stats, N);

  gemm_bn_kernel<<<gmm, 256, 0, stream>>>(tbuf, w2, hbuf, stats + 2 * HID, N);
  bn_relu_kernel<<<2048, 256, 0, stream>>>(hbuf, g2, b2, stats + 2 * HID, N);

  dim3 gmo(Mpad / MBLK, H / NBLK);
  gemm_out_kernel<<<gmo, 256, 0, stream>>>(hbuf, w3, b3, out, N);
}